// MMCL_32289564131845
// MI455X (gfx1250) — compile-verified
//
#include <hip/hip_runtime.h>
#include <hip/hip_bf16.h>
#include <math.h>

// ---------------------------------------------------------------------------
// MMCL loss on MI455X (gfx1250), one workgroup (256 thr = 8 wave32) per row.
//   Stage 1: Tensor Data Mover DMAs the 128KB fp32 row HBM->LDS (8 chunks,
//            one per wave), s_wait_tensorcnt + barrier.   [single HBM read]
//   Pass A : fused row-max + top-byte histogram (per-wave privatized bins).
//   Pass B : fused exp-sum over buckets > b1  +  compaction of bucket-b1
//            candidates (~1% of row) into LDS.
//   Tail   : bitonic sort of candidates (deterministic), exp-sum of top k1,
//            assemble logsumexp.  Fallback to full radix select on overflow.
// Only 2 full LDS sweeps after the DMA -> HBM-bandwidth-bound (~11us floor).
// ---------------------------------------------------------------------------

typedef __attribute__((ext_vector_type(4))) unsigned int uint32x4_t_;
typedef __attribute__((ext_vector_type(8))) int          int32x8_t_;
typedef __attribute__((ext_vector_type(4))) int          int32x4_t_;

#define TPB     256
#define NWAVE   (TPB / 32)       // wave32 on CDNA5
#define CAND_CAP 4096            // candidate buffer (power of two, for sort)
#define NEGINF_F (-1e30f)

// Monotone mapping: float -> uint32 key, descending float == descending key.
__device__ __forceinline__ unsigned f2key(float f) {
    unsigned u = __float_as_uint(f);
    return u ^ ((unsigned)((int)u >> 31) | 0x80000000u);
}
__device__ __forceinline__ float key2f(unsigned k) {
    unsigned u = (k & 0x80000000u) ? (k ^ 0x80000000u) : ~k;
    return __uint_as_float(u);
}

__global__ void mmcl_row_kernel(const float* __restrict__ logits,
                                const int*  __restrict__ targets,
                                float* __restrict__ row_loss,
                                int N, int K, int use_tdm)
{
    extern __shared__ __align__(16) char smem[];
    float*    row  = (float*)smem;                                     // N floats
    unsigned* hist = (unsigned*)(smem + (size_t)N * sizeof(float));    // NWAVE*256
    unsigned* cand = hist + NWAVE * 256;                               // CAND_CAP

    __shared__ unsigned s_hist[256];
    __shared__ float    s_red[TPB];
    __shared__ float    s_pos, s_v1, s_sumgt;
    __shared__ unsigned s_pref, s_cnt, s_m2;
    __shared__ int      s_k;

    const int tid    = threadIdx.x;
    const int rowIdx = blockIdx.x;
    const int w      = tid >> 5;                 // wave id (wave32)
    const float* g   = logits + (size_t)rowIdx * (size_t)N;
    const int vec    = (N % (TPB * 4)) == 0;

    // ---- Stage 1: row HBM -> LDS via Tensor Data Mover -----------------
    if (use_tdm) {
        const int CH = N / NWAVE;                // fp32 elements per wave DMA
        unsigned lds_off = (unsigned)(size_t)(const void*)(row + (size_t)w * CH);
        unsigned long long ga =
            (unsigned long long)(size_t)(const void*)(g + (size_t)w * CH);

        // D# group 0: count=1, lds_addr, 57-bit global_addr, type=2 ("image")
        uint32x4_t_ g0 = {
            1u,
            lds_off,
            (unsigned)(ga & 0xFFFFFFFFull),
            (unsigned)((ga >> 32) & 0x01FFFFFFull) | (2u << 30)
        };
        // D# group 1: data_size=4B; 1-D tensor/tile of CH elements, dim1=1.
        int32x8_t_ g1 = {
            (int)(2u << 16),                                       // data_size=2 (4B)
            (int)(((unsigned)CH & 0xFFFFu) << 16),                 // tensor_dim0[15:0]
            (int)((((unsigned)CH >> 16) & 0xFFFFu) | (1u << 16)),  // dim0 hi | tensor_dim1=1
            (int)(((unsigned)CH & 0xFFFFu) << 16),                 // tile_dim0 = CH
            1,                                                     // tile_dim1 = 1
            CH,                                                    // tensor_dim0_stride
            0, 0
        };
        int32x4_t_ zz4 = {0, 0, 0, 0};
        int32x8_t_ zz8 = {0, 0, 0, 0, 0, 0, 0, 0};
        __builtin_amdgcn_tensor_load_to_lds(g0, g1, zz4, zz4, zz8, 0);
        __builtin_amdgcn_s_wait_tensorcnt(0);    // wait this wave's DMA
    } else {
        for (int i = tid; i < N; i += TPB) row[i] = g[i];
    }
    __syncthreads();

    // ---- Stage 2: extract positive, mask it, init state -----------------
    if (tid == 0) {
        int t  = targets[rowIdx];
        s_pos  = row[t];
        row[t] = NEGINF_F;           // matches reference NEG_INF masking
        s_pref = 0u;
        s_k    = K;
        s_cnt  = 0u;
    }
    for (int i = tid; i < NWAVE * 256; i += TPB) hist[i] = 0u;
    __syncthreads();

    // ---- Pass A: fused row-max + top-byte histogram ----------------------
    {
        unsigned* myhist = hist + w * 256;
        float lmax = NEGINF_F;
        if (vec) {
            const float4* row4 = (const float4*)row;
            for (int i = tid; i < (N >> 2); i += TPB) {
                float4 v = row4[i];
                atomicAdd(&myhist[f2key(v.x) >> 24], 1u); lmax = fmaxf(lmax, v.x);
                atomicAdd(&myhist[f2key(v.y) >> 24], 1u); lmax = fmaxf(lmax, v.y);
                atomicAdd(&myhist[f2key(v.z) >> 24], 1u); lmax = fmaxf(lmax, v.z);
                atomicAdd(&myhist[f2key(v.w) >> 24], 1u); lmax = fmaxf(lmax, v.w);
            }
        } else {
            for (int i = tid; i < N; i += TPB) {
                float v = row[i];
                atomicAdd(&myhist[f2key(v) >> 24], 1u); lmax = fmaxf(lmax, v);
            }
        }
        s_red[tid] = lmax;
        __syncthreads();
        // combine per-wave histograms (one bucket per thread; TPB == 256)
        unsigned tot = 0u;
        for (int w2 = 0; w2 < NWAVE; ++w2) tot += hist[w2 * 256 + tid];
        s_hist[tid] = tot;
        // deterministic max tree-reduce (syncs also publish s_hist)
        for (int s = TPB / 2; s > 0; s >>= 1) {
            __syncthreads();
            if (tid < s) s_red[tid] = fmaxf(s_red[tid], s_red[tid + s]);
        }
        __syncthreads();
        if (tid == 0) {
            s_v1 = s_red[0];
            int kk = s_k, cum = 0, b = 255;
            for (; b > 0; --b) {
                int c = (int)s_hist[b];
                if (cum + c >= kk) break;
                cum += c;
            }
            s_k    = kk - cum;                   // k1: rank inside bucket b1
            s_pref = (s_pref << 8) | (unsigned)b;
        }
        __syncthreads();
    }
    const float    v1  = s_v1;
    const float    pos = s_pos;
    const unsigned b1  = s_pref & 0xFFu;
    const int      k1  = s_k;
    const float    M   = 10.0f * fmaxf(pos, v1);

    // ---- Pass B: exp-sum of buckets > b1  +  compact bucket-b1 candidates
    {
        float lsum = 0.0f;
        if (vec) {
            const float4* row4 = (const float4*)row;
            for (int i = tid; i < (N >> 2); i += TPB) {
                float4 q = row4[i];
                float vv[4] = {q.x, q.y, q.z, q.w};
                #pragma unroll
                for (int u = 0; u < 4; ++u) {
                    unsigned key = f2key(vv[u]);
                    unsigned tb  = key >> 24;
                    if (tb > b1) {
                        lsum += expf(10.0f * vv[u] - M);
                    } else if (tb == b1) {
                        unsigned idx = atomicAdd(&s_cnt, 1u);
                        if (idx < CAND_CAP) cand[idx] = key;
                    }
                }
            }
        } else {
            for (int i = tid; i < N; i += TPB) {
                float v = row[i];
                unsigned key = f2key(v);
                unsigned tb  = key >> 24;
                if (tb > b1) {
                    lsum += expf(10.0f * v - M);
                } else if (tb == b1) {
                    unsigned idx = atomicAdd(&s_cnt, 1u);
                    if (idx < CAND_CAP) cand[idx] = key;
                }
            }
        }
        s_red[tid] = lsum;
        for (int s = TPB / 2; s > 0; s >>= 1) {
            __syncthreads();
            if (tid < s) s_red[tid] += s_red[tid + s];
        }
        __syncthreads();
        if (tid == 0) s_sumgt = s_red[0];
        __syncthreads();
    }
    const unsigned cnt = s_cnt;

    if (cnt <= CAND_CAP) {
        // ---- Fast path: bitonic-sort the tiny candidate set (descending) --
        if (tid == 0) {
            unsigned m = 2u;
            while (m < cnt) m <<= 1;
            s_m2 = m;
        }
        __syncthreads();
        const unsigned m2 = s_m2;
        for (unsigned i = cnt + tid; i < m2; i += TPB) cand[i] = 0u;  // pad = -min key
        __syncthreads();
        for (unsigned kk = 2; kk <= m2; kk <<= 1) {
            for (unsigned j = kk >> 1; j > 0; j >>= 1) {
                for (unsigned i = tid; i < m2; i += TPB) {
                    unsigned l = i ^ j;
                    if (l > i) {
                        unsigned a = cand[i], b = cand[l];
                        bool descBlock = ((i & kk) == 0);
                        if (descBlock ? (a < b) : (a > b)) {
                            cand[i] = b; cand[l] = a;
                        }
                    }
                }
                __syncthreads();
            }
        }
        // exp-sum of the top k1 candidates (sorted order -> deterministic)
        float csum = 0.0f;
        for (int i = tid; i < k1; i += TPB)
            csum += expf(10.0f * key2f(cand[i]) - M);
        s_red[tid] = csum;
        for (int s = TPB / 2; s > 0; s >>= 1) {
            __syncthreads();
            if (tid < s) s_red[tid] += s_red[tid + s];
        }
        __syncthreads();
        if (tid == 0) {
            float S = s_sumgt + s_red[0]
                    - expf(10.0f * v1  - M)     // drop hardest negative
                    + expf(10.0f * pos - M);    // the positive (class 0)
            row_loss[rowIdx] = logf(S) + M - 10.0f * pos;
        }
    } else {
        // ---- Fallback: finish radix select over the full row (d = 2..0) --
        for (int d = 2; d >= 0; --d) {
            for (int i = tid; i < NWAVE * 256; i += TPB) hist[i] = 0u;
            __syncthreads();
            const unsigned pref  = s_pref;
            const int      shift = d * 8;
            for (int i = tid; i < N; i += TPB) {
                unsigned key = f2key(row[i]);
                if ((key >> (shift + 8)) == pref)
                    atomicAdd(&hist[w * 256 + ((key >> shift) & 0xFFu)], 1u);
            }
            __syncthreads();
            unsigned tot = 0u;
            for (int w2 = 0; w2 < NWAVE; ++w2) tot += hist[w2 * 256 + tid];
            s_hist[tid] = tot;
            __syncthreads();
            if (tid == 0) {
                int kk = s_k, cum = 0, b = 255;
                for (; b > 0; --b) {
                    int c = (int)s_hist[b];
                    if (cum + c >= kk) break;
                    cum += c;
                }
                s_k    = kk - cum;
                s_pref = (s_pref << 8) | (unsigned)b;
            }
            __syncthreads();
        }
        const unsigned tauKey = s_pref;
        const int      ktie   = s_k;
        const float    tau    = key2f(tauKey);
        float lsum = 0.0f;
        for (int i = tid; i < N; i += TPB) {
            float v = row[i];
            if (f2key(v) > tauKey) lsum += expf(10.0f * v - M);
        }
        s_red[tid] = lsum;
        for (int s = TPB / 2; s > 0; s >>= 1) {
            __syncthreads();
            if (tid < s) s_red[tid] += s_red[tid + s];
        }
        __syncthreads();
        if (tid == 0) {
            float S = s_red[0]
                    + (float)ktie * expf(10.0f * tau - M)
                    - expf(10.0f * v1  - M)
                    + expf(10.0f * pos - M);
            row_loss[rowIdx] = logf(S) + M - 10.0f * pos;
        }
    }
}

// Deterministic single-block mean over B per-row losses.
__global__ void mmcl_reduce_kernel(const float* __restrict__ row_loss,
                                   float* __restrict__ out, int B)
{
    __shared__ float red[TPB];
    float s = 0.0f;
    for (int i = threadIdx.x; i < B; i += TPB) s += row_loss[i];
    red[threadIdx.x] = s;
    __syncthreads();
    for (int st = TPB / 2; st > 0; st >>= 1) {
        if (threadIdx.x < st) red[threadIdx.x] += red[threadIdx.x + st];
        __syncthreads();
    }
    if (threadIdx.x == 0) out[0] = red[0] / (float)B;
}

extern "C" void kernel_launch(void* const* d_in, const int* in_sizes, int n_in,
                              void* d_out, int out_size, void* d_ws, size_t ws_size,
                              hipStream_t stream) {
    const float* logits  = (const float*)d_in[0];
    const int*   targets = (const int*)d_in[1];

    const int B = in_sizes[1];
    const int N = in_sizes[0] / B;
    const int K = (int)(0.01 * (double)(N - 1));   // = 327 for N=32768

    float* row_loss = (float*)d_ws;                 // B floats of scratch

    // Dynamic LDS: row buffer + per-wave histograms + candidate buffer.
    const size_t smem = (size_t)N * sizeof(float)
                      + (size_t)NWAVE * 256 * sizeof(unsigned)
                      + (size_t)CAND_CAP * sizeof(unsigned);

    // TDM path needs equal per-wave chunks with tile_dim0 fitting 16 bits.
    const int ch      = N / NWAVE;
    const int use_tdm = ((N % NWAVE) == 0) && (ch > 0) && (ch <= 0xFFFF);

    mmcl_row_kernel<<<B, TPB, smem, stream>>>(logits, targets, row_loss, N, K, use_tdm);
    mmcl_reduce_kernel<<<1, TPB, 0, stream>>>(row_loss, (float*)d_out, B);
}